// ForwardProjectionLite_16097537425502
// MI455X (gfx1250) — compile-verified
//
#include <hip/hip_runtime.h>
#include <hip/hip_bf16.h>

// ---------------------------------------------------------------------------
// BEV forward projection (Lift-Splat) for MI455X / gfx1250, wave32.
//   B=1, NCAM=6, C=80, D=88, HF=32, WF=88, BEV 128x128x8
// Kernel 1: fold camera transforms (6 cameras) -> workspace
// Kernel 2: one wave per (n,h,w) pixel:
//   - compute 88 voxel output-offsets into LDS, hoist into regs per d-block
//   - lift = depth(16) x ctx(16) outer products via V_WMMA_F32_16X16X4_F32
//   - scatter rows with global_atomic_add_f32 (saddr + 32-bit voffset form)
// ---------------------------------------------------------------------------

#define NCAM 6
#define CCH  80
#define DD   88
#define HF   32
#define WF   88
#define BEV_H 128
#define BEV_W 128
#define BEV_Z 8
// output strides for [C, Hbev, Wbev, Zbev]
#define OSTRIDE_C (BEV_H * BEV_W * BEV_Z)   // 131072

typedef __attribute__((ext_vector_type(2))) float v2f;
typedef __attribute__((ext_vector_type(8))) float v8f;

// ---- 3x3 helpers -----------------------------------------------------------
__device__ inline void inv3x3(const float* m, float* o) {
    float det = m[0] * (m[4] * m[8] - m[5] * m[7])
              - m[1] * (m[3] * m[8] - m[5] * m[6])
              + m[2] * (m[3] * m[7] - m[4] * m[6]);
    float id = 1.0f / det;
    o[0] = (m[4] * m[8] - m[5] * m[7]) * id;
    o[1] = (m[2] * m[7] - m[1] * m[8]) * id;
    o[2] = (m[1] * m[5] - m[2] * m[4]) * id;
    o[3] = (m[5] * m[6] - m[3] * m[8]) * id;
    o[4] = (m[0] * m[8] - m[2] * m[6]) * id;
    o[5] = (m[2] * m[3] - m[0] * m[5]) * id;
    o[6] = (m[3] * m[7] - m[4] * m[6]) * id;
    o[7] = (m[1] * m[6] - m[0] * m[7]) * id;
    o[8] = (m[0] * m[4] - m[1] * m[3]) * id;
}
__device__ inline void mm3x3(const float* a, const float* b, float* o) {
    for (int i = 0; i < 3; ++i)
        for (int j = 0; j < 3; ++j)
            o[i * 3 + j] = a[i * 3 + 0] * b[0 * 3 + j]
                         + a[i * 3 + 1] * b[1 * 3 + j]
                         + a[i * 3 + 2] * b[2 * 3 + j];
}

// ---- Kernel 1: per-camera fold --------------------------------------------
__global__ void fp_prep_kernel(const float* __restrict__ rots,
                               const float* __restrict__ trans,
                               const float* __restrict__ intrins,
                               const float* __restrict__ post_rots,
                               const float* __restrict__ post_trans,
                               const float* __restrict__ bda,
                               float* __restrict__ fold) {
    int n = threadIdx.x;
    if (n >= NCAM) return;
    float R[9], I[9], P[9], Bd[9], invI[9], invP[9], RI[9], M[9];
    for (int i = 0; i < 9; ++i) {
        R[i]  = rots[n * 9 + i];
        I[i]  = intrins[n * 9 + i];
        P[i]  = post_rots[n * 9 + i];
        Bd[i] = bda[i];
    }
    inv3x3(I, invI);
    inv3x3(P, invP);
    mm3x3(R, invI, RI);
    mm3x3(Bd, RI, M);
    float* w = fold + n * 24;
    for (int i = 0; i < 9; ++i) w[i] = M[i];
    float tx = trans[n * 3 + 0], ty = trans[n * 3 + 1], tz = trans[n * 3 + 2];
    w[ 9] = Bd[0] * tx + Bd[1] * ty + Bd[2] * tz;
    w[10] = Bd[3] * tx + Bd[4] * ty + Bd[5] * tz;
    w[11] = Bd[6] * tx + Bd[7] * ty + Bd[8] * tz;
    for (int i = 0; i < 9; ++i) w[12 + i] = invP[i];
    w[21] = post_trans[n * 3 + 0];
    w[22] = post_trans[n * 3 + 1];
    w[23] = post_trans[n * 3 + 2];
}

// ---- Kernel 2: main lift + splat ------------------------------------------
__global__ void __launch_bounds__(32)
fp_splat_kernel(const float* __restrict__ ctx,   // [N,C,HF,WF]
                const float* __restrict__ dp,    // [N,D,HF,WF]
                const float* __restrict__ fold,  // [N,24]
                float* __restrict__ out) {       // [C,128,128,8]
    const int pix = blockIdx.x;
    const int n = pix / (HF * WF);
    const int rem = pix - n * (HF * WF);
    const int h = rem / WF;
    const int w = rem - h * WF;
    const int lane = threadIdx.x;                // 0..31 (wave32)
    const int lane16 = lane & 15;
    const bool lo = lane < 16;

    const float* F = fold + n * 24;
    float M0 = F[0], M1 = F[1], M2 = F[2];
    float M3 = F[3], M4 = F[4], M5 = F[5];
    float M6 = F[6], M7 = F[7], M8 = F[8];
    float T0 = F[9], T1 = F[10], T2 = F[11];
    float P0 = F[12], P1 = F[13], P2 = F[14];
    float P3 = F[15], P4 = F[16], P5 = F[17];
    float P6 = F[18], P7 = F[19], P8 = F[20];
    float pT0 = F[21], pT1 = F[22], pT2 = F[23];

    __shared__ int offs[96];
    const float step = (61.0f - 2.0f) / (float)(DD - 1);
    for (int d = lane; d < 96; d += 32) {
        int o = -1;
        if (d < DD) {
            float dep = 2.0f + (float)d * step;
            float px = (float)w - pT0;
            float py = (float)h - pT1;
            float pz = dep - pT2;
            float qx = P0 * px + P1 * py + P2 * pz;
            float qy = P3 * px + P4 * py + P5 * pz;
            float qz = P6 * px + P7 * py + P8 * pz;
            float rx = qx * qz, ry = qy * qz, rz = qz;
            float sx = M0 * rx + M1 * ry + M2 * rz + T0;
            float sy = M3 * rx + M4 * ry + M5 * rz + T1;
            float sz = M6 * rx + M7 * ry + M8 * rz + T2;
            int xi = (int)((sx + 51.2f) / 0.8f);
            int yi = (int)((sy + 51.2f) / 0.8f);
            int zi = (int)((sz + 5.0f) / 1.0f);
            if (xi >= 0 && xi < BEV_W && yi >= 0 && yi < BEV_H &&
                zi >= 0 && zi < BEV_Z)
                o = (yi * BEV_W + xi) * BEV_Z + zi;
        }
        offs[d] = o;
    }
    __syncthreads();

    const int pixoff = h * WF + w;

    float ctxv[5];
#pragma unroll
    for (int cb = 0; cb < 5; ++cb) {
        float v = 0.0f;
        if (lo) v = ctx[((n * CCH + cb * 16 + lane16) * HF * WF) + pixoff];
        ctxv[cb] = v;
    }
    float dpv[6];
#pragma unroll
    for (int db = 0; db < 6; ++db) {
        float v = 0.0f;
        int didx = db * 16 + lane16;
        if (lo && didx < DD) v = dp[((n * DD + didx) * HF * WF) + pixoff];
        dpv[db] = v;
    }

    const int rbase = (lo ? 0 : 8);
#pragma unroll
    for (int db = 0; db < 6; ++db) {
        int offr[8];
#pragma unroll
        for (int r = 0; r < 8; ++r) offr[r] = offs[db * 16 + rbase + r];

        v2f a;
        a.x = dpv[db];
        a.y = 0.0f;
#pragma unroll
        for (int cb = 0; cb < 5; ++cb) {
            v2f b;
            b.x = lo ? ctxv[cb] : 0.0f;
            b.y = 0.0f;
            v8f acc = {};
            acc = __builtin_amdgcn_wmma_f32_16x16x4_f32(
                false, a, false, b, (short)0, acc, false, false);
            const unsigned cbase = (unsigned)(cb * 16 + lane16) * OSTRIDE_C;
#pragma unroll
            for (int r = 0; r < 8; ++r) {
                int off = offr[r];
                if (off >= 0) {
                    unsigned idx = cbase + (unsigned)off;
                    atomicAdd(&out[idx], acc[r]);
                }
            }
        }
    }
}

// ---------------------------------------------------------------------------
extern "C" void kernel_launch(void* const* d_in, const int* in_sizes, int n_in,
                              void* d_out, int out_size, void* d_ws, size_t ws_size,
                              hipStream_t stream) {
    const float* ctx        = (const float*)d_in[0];
    const float* dp         = (const float*)d_in[1];
    const float* rots       = (const float*)d_in[2];
    const float* trans      = (const float*)d_in[3];
    const float* intrins    = (const float*)d_in[4];
    const float* post_rots  = (const float*)d_in[5];
    const float* post_trans = (const float*)d_in[6];
    const float* bda        = (const float*)d_in[7];
    float* out = (float*)d_out;
    float* fold = (float*)d_ws;

    hipMemsetAsync(out, 0, (size_t)out_size * sizeof(float), stream);

    fp_prep_kernel<<<1, 32, 0, stream>>>(rots, trans, intrins, post_rots,
                                         post_trans, bda, fold);

    fp_splat_kernel<<<NCAM * HF * WF, 32, 0, stream>>>(ctx, dp, fold, out);
}